// MoE_11785390260960
// MI455X (gfx1250) — compile-verified
//
#include <hip/hip_runtime.h>
#include <math.h>

typedef __attribute__((ext_vector_type(2))) float v2f;
typedef __attribute__((ext_vector_type(8))) float v8f;

#define TOKENS 4096
#define DDIM   1024
#define FDIM   2048
#define C2F    4096   // 2*F
#define NEXP   8
#define TM     64     // tokens per CTA tile (big tile -> half the L2 weight traffic)
#define NTHR   512    // 16 waves
#define XSTR   1028   // X LDS row stride (floats), 16B aligned rows, bank-skewed
#define HSTR   66     // Hact LDS row stride
#define SSTR   33     // stage row stride

__device__ __forceinline__ v8f wmma4(v2f a, v2f b, v8f c) {
  // V_WMMA_F32_16X16X4_F32 : D = A(16x4) * B(4x16) + C(16x16)
  return __builtin_amdgcn_wmma_f32_16x16x4_f32(
      /*neg_a=*/false, a, /*neg_b=*/false, b,
      /*c_mod=*/(short)0, c, /*reuse_a=*/false, /*reuse_b=*/false);
}

// ---------------------------------------------------------------- zero kernel
__global__ void k_zero(float* __restrict__ out, int* __restrict__ counts) {
  int idx = blockIdx.x * blockDim.x + threadIdx.x;
  if (idx < NEXP) counts[idx] = 0;
  int stride = gridDim.x * blockDim.x;
  for (int i = idx; i < TOKENS * DDIM; i += stride) out[i] = 0.0f;
}

// ----------------------------------------------------- gate + top2 + routing
__global__ void __launch_bounds__(256)
k_gate(const float* __restrict__ x, const float* __restrict__ gw,
       int* __restrict__ counts, int* __restrict__ tokids,
       float* __restrict__ tokw) {
  const int tid  = threadIdx.x;
  const int wave = tid >> 5;
  const int lane = tid & 31;
  const int t    = blockIdx.x * 8 + wave;   // one token per wave

  float acc[NEXP];
#pragma unroll
  for (int e = 0; e < NEXP; ++e) acc[e] = 0.0f;

  const float* xr = x + (size_t)t * DDIM;
  for (int d = lane; d < DDIM; d += 32) {
    float xv = xr[d];
    const float* gr = gw + (size_t)d * NEXP;
#pragma unroll
    for (int e = 0; e < NEXP; ++e) acc[e] += xv * gr[e];
  }
  // cross-lane reduction (wave32)
#pragma unroll
  for (int e = 0; e < NEXP; ++e) {
#pragma unroll
    for (int off = 16; off > 0; off >>= 1)
      acc[e] += __shfl_xor(acc[e], off, 32);
  }

  if (lane == 0) {
    float g0 = -INFINITY, g1 = -INFINITY;
    int   i0 = 0, i1 = 0;
#pragma unroll
    for (int e = 0; e < NEXP; ++e) {
      float v = acc[e];
      if (v > g0)      { g1 = g0; i1 = i0; g0 = v; i0 = e; }
      else if (v > g1) { g1 = v;  i1 = e; }
    }
    float e1 = __expf(g1 - g0);        // softmax over the two
    float w0 = 1.0f / (1.0f + e1);
    float w1 = e1 * w0;
    int p0 = atomicAdd(&counts[i0], 1);
    tokids[i0 * TOKENS + p0] = t;
    tokw  [i0 * TOKENS + p0] = w0;
    int p1 = atomicAdd(&counts[i1], 1);
    tokids[i1 * TOKENS + p1] = t;
    tokw  [i1 * TOKENS + p1] = w1;
  }
}

// ------------------------------------------------ fused expert FFN (WMMA f32)
__global__ void __launch_bounds__(NTHR)
k_expert(const float* __restrict__ x,  const float* __restrict__ w1,
         const float* __restrict__ b1, const float* __restrict__ w2,
         const float* __restrict__ b2, const int* __restrict__ counts,
         const int* __restrict__ tokids, const float* __restrict__ tokw,
         float* __restrict__ out) {
  __shared__ float Xlds [TM * XSTR];         // 263168 B : 64 x 1024 token tile
  __shared__ float Hact [TM * HSTR];         //  16896 B : 64 x 64 swiglu chunk
  __shared__ float Stage[16 * 16 * SSTR];    //  33792 B : per-wave Hraw staging
  __shared__ int   TokId[TM];
  __shared__ float TokW [TM];

  const int e  = blockIdx.y;
  const int nt = counts[e];
  const int t0 = blockIdx.x * TM;
  if (t0 >= nt) return;
  const int ntile = min(TM, nt - t0);

  const int tid = threadIdx.x;
  if (tid < TM) {
    if (tid < ntile) {
      TokId[tid] = tokids[e * TOKENS + t0 + tid];
      TokW [tid] = tokw  [e * TOKENS + t0 + tid];
    } else { TokId[tid] = -1; TokW[tid] = 0.0f; }
  }
  __syncthreads();

  // stage X tile (zero-padded rows), float4 vectorized
  for (int idx = tid; idx < TM * (DDIM / 4); idx += NTHR) {
    int row = idx >> 8;            // DDIM/4 == 256
    int c4  = idx & 255;
    float4 v = make_float4(0.f, 0.f, 0.f, 0.f);
    int tk = TokId[row];
    if (tk >= 0) v = ((const float4*)(x + (size_t)tk * DDIM))[c4];
    *(float4*)(&Xlds[row * XSTR + c4 * 4]) = v;
  }
  __syncthreads();

  const int wave    = tid >> 5;
  const int lane    = tid & 31;
  const int halfsel = lane >> 4;        // 0: lanes 0-15, 1: lanes 16-31
  const int nlo     = lane & 15;
  const int kofs    = halfsel << 1;     // K offset within 4-wide step
  const int mt      = wave & 3;         // token quarter (0..3)
  const int grp     = wave >> 2;        // column group (0..3)

  const float* w1e = w1 + (size_t)e * C2F * DDIM;
  const float* w2e = w2 + (size_t)e * DDIM * FDIM;

  v8f acc[16];                          // Y tile: 16 tokens x 256 d-cols / wave
#pragma unroll
  for (int j = 0; j < 16; ++j) acc[j] = (v8f){0,0,0,0,0,0,0,0};

  const int arow = mt * 16 + nlo;       // A-fragment row (token within tile)

  for (int f0 = 0; f0 < FDIM; f0 += 64) {
    // hint: upcoming W2 chunk (emits global_prefetch_b8)
    __builtin_prefetch(w2e + (size_t)(grp * 256 + nlo) * FDIM + f0, 0, 1);

    // ---- GEMM1: Hraw[64 x 128c] for c in [2*f0, 2*f0+128), K = D = 1024
    const int cstart = 2 * f0 + grp * 32;
    v8f c0 = (v8f){0,0,0,0,0,0,0,0};
    v8f c1 = (v8f){0,0,0,0,0,0,0,0};
    const float* w1r0 = w1e + (size_t)(cstart + nlo)      * DDIM;
    const float* w1r1 = w1e + (size_t)(cstart + 16 + nlo) * DDIM;
#pragma unroll 4
    for (int d = 0; d < DDIM; d += 4) {
      v2f a   = *(const v2f*)(&Xlds[arow * XSTR + d + kofs]);
      v2f bb0 = *(const v2f*)(w1r0 + d + kofs);
      v2f bb1 = *(const v2f*)(w1r1 + d + kofs);
      c0 = wmma4(a, bb0, c0);
      c1 = wmma4(a, bb1, c1);
    }
    // bias + stage Hraw to LDS for cross-lane swiglu pairing
    float bv0 = b1[(size_t)e * C2F + cstart + nlo];
    float bv1 = b1[(size_t)e * C2F + cstart + 16 + nlo];
#pragma unroll
    for (int i = 0; i < 8; ++i) {
      int r = i + 8 * halfsel;
      Stage[(wave * 16 + r) * SSTR + nlo]      = c0[i] + bv0;
      Stage[(wave * 16 + r) * SSTR + 16 + nlo] = c1[i] + bv1;
    }
    __syncthreads();

    // ---- swiglu: Hact[tok][f_local], tok in [0,64), f_local in [0,64)
#pragma unroll
    for (int it = 0; it < 8; ++it) {
      int idx = tid + it * NTHR;         // 4096 elements
      int tok = idx >> 6;
      int fl  = idx & 63;
      int sw  = (tok >> 4) | ((fl >> 4) << 2);   // source wave (mt | grp<<2)
      int r   = tok & 15;
      int cl  = (fl & 15) << 1;
      float g = Stage[(sw * 16 + r) * SSTR + cl];
      float v = Stage[(sw * 16 + r) * SSTR + cl + 1];
      g = fminf(g, 9.0f);
      v = fminf(fmaxf(v, -9.0f), 9.0f);
      float s = 1.0f / (1.0f + __expf(-1.702f * g));
      Hact[tok * HSTR + fl] = g * s * (v + 1.0f);
    }
    __syncthreads();

    // ---- GEMM2 partial: Y[64 x 1024] += Hact(64x64) @ W2^T chunk, K = 64
    const float* w2f = w2e + f0;
#pragma unroll 2
    for (int kk = 0; kk < 64; kk += 4) {
      v2f a = *(const v2f*)(&Hact[arow * HSTR + kk + kofs]);
#pragma unroll
      for (int j = 0; j < 16; ++j) {
        int n = grp * 256 + j * 16 + nlo;
        v2f b = *(const v2f*)(w2f + (size_t)n * FDIM + kk + kofs);
        acc[j] = wmma4(a, b, acc[j]);
      }
    }
    __syncthreads();   // Stage/Hact reused next chunk
  }

  // ---- epilogue: += routing_weight * (Y + b2)  (scatter-add, top-2 overlap)
#pragma unroll
  for (int j = 0; j < 16; ++j) {
    int n = grp * 256 + j * 16 + nlo;
    float b2v = b2[(size_t)e * DDIM + n];
#pragma unroll
    for (int i = 0; i < 8; ++i) {
      int r = mt * 16 + i + 8 * halfsel;
      if (r < ntile) {
        int   tk = TokId[r];
        float wt = TokW[r];
        atomicAdd(&out[(size_t)tk * DDIM + n], wt * (acc[j][i] + b2v));
      }
    }
  }
}

// ----------------------------------------------------------------- launcher
extern "C" void kernel_launch(void* const* d_in, const int* in_sizes, int n_in,
                              void* d_out, int out_size, void* d_ws, size_t ws_size,
                              hipStream_t stream) {
  const float* x  = (const float*)d_in[0];
  const float* gw = (const float*)d_in[1];
  const float* w1 = (const float*)d_in[2];
  const float* b1 = (const float*)d_in[3];
  const float* w2 = (const float*)d_in[4];
  const float* b2 = (const float*)d_in[5];
  float* out = (float*)d_out;

  char*  ws     = (char*)d_ws;
  int*   counts = (int*)ws;                               // 8 ints (256B pad)
  int*   tokids = (int*)(ws + 256);                       // 8*4096 ints
  float* tokw   = (float*)(ws + 256 + NEXP * TOKENS * 4); // 8*4096 floats

  k_zero<<<1024, 256, 0, stream>>>(out, counts);
  k_gate<<<TOKENS / 8, 256, 0, stream>>>(x, gw, counts, tokids, tokw);
  dim3 grid(TOKENS / TM, NEXP);
  k_expert<<<grid, NTHR, 0, stream>>>(x, w1, b1, w2, b2, counts, tokids, tokw, out);
}